// GCNLayer_35253091566190
// MI455X (gfx1250) — compile-verified
//
#include <hip/hip_runtime.h>

#define N_NODES 100000
#define N_EDGES 1600000
#define D_FEAT  64

typedef unsigned int uint32x4 __attribute__((ext_vector_type(4)));
typedef int          int32x4  __attribute__((ext_vector_type(4)));
typedef int          int32x8  __attribute__((ext_vector_type(8)));

// Kernel 1: zero the output (d_out is poisoned; scatter accumulates atomically).
__global__ __launch_bounds__(256) void zero_out_kernel(float4* __restrict__ out, int n4) {
    int i = blockIdx.x * blockDim.x + threadIdx.x;
    if (i < n4) out[i] = make_float4(0.f, 0.f, 0.f, 0.f);
}

// COO scatter-add with TDM gather-mode row fetch.
// One wave32 owns 32 edges. The 32 col indices are readlane'd into SGPRs and
// fed to the Tensor Data Mover in gather mode (8 x 32-bit row indices per
// descriptor): the TDM DMAs 8 embedding rows (8 x 256B) into this wave's LDS
// buffer, tracked by TENSORcnt. Two buffers are double-buffered so one 2KB
// gather is always in flight while the wave scales the previous 8 rows and
// fires non-returning global_atomic_add_f32 scatters (resolved at L2).
__global__ __launch_bounds__(256) void spmm_coo_scatter_tdm(
    const int*   __restrict__ rows,
    const int*   __restrict__ cols,
    const float* __restrict__ vals,
    const float* __restrict__ embeds,
    float*       __restrict__ out,
    int E)
{
    extern __shared__ float smem[];   // 8 waves * 2 bufs * 8 rows * 64 f32 = 32KB

    const int lane = threadIdx.x & 31;
    const int wave = (blockIdx.x * blockDim.x + threadIdx.x) >> 5;
    const int base = wave * 32;
    if (base >= E) return;

    int   r = 0, c = 0;
    float v = 0.0f;
    const int e = base + lane;
    if (e < E) {
        r = __builtin_nontemporal_load(rows + e);   // edge stream: read once, NT
        c = __builtin_nontemporal_load(cols + e);
        v = __builtin_nontemporal_load(vals + e);
    }

    const int feat = lane * 2;                      // float2 per lane = 256B/row

    if (base + 32 <= E) {
        // ---- fast path: full 32-edge block (always taken for E = 1.6M) ----
        const int warp_in_blk =
            __builtin_amdgcn_readfirstlane((int)(threadIdx.x >> 5));
        // float-index base of this wave's LDS slot (1024 floats = 4KB)
        const int slot = warp_in_blk * 1024;
        // byte offsets of the two 2KB gather buffers (group-segment relative;
        // smem is the only LDS object, dynamic LDS starts at offset 0)
        const unsigned buf_bytes[2] = { (unsigned)(slot * 4),
                                        (unsigned)(slot * 4 + 2048) };
        const unsigned long long gaddr = (unsigned long long)(uintptr_t)embeds;

        // Issue one gather-mode TDM op for edges [8K, 8K+8) into buffer BUF.
#define TDM_ISSUE(K, BUF)                                                      \
        do {                                                                   \
            int32x4 g2 = { __builtin_amdgcn_readlane(c, (K)*8 + 0),            \
                           __builtin_amdgcn_readlane(c, (K)*8 + 1),            \
                           __builtin_amdgcn_readlane(c, (K)*8 + 2),            \
                           __builtin_amdgcn_readlane(c, (K)*8 + 3) };          \
            int32x4 g3 = { __builtin_amdgcn_readlane(c, (K)*8 + 4),            \
                           __builtin_amdgcn_readlane(c, (K)*8 + 5),            \
                           __builtin_amdgcn_readlane(c, (K)*8 + 6),            \
                           __builtin_amdgcn_readlane(c, (K)*8 + 7) };          \
            uint32x4 g0 = {                                                    \
                1u | (1u << 30) | (1u << 31),      /* count=1, 32b idx, gather */\
                buf_bytes[BUF],                    /* lds_addr               */ \
                (unsigned)(gaddr & 0xFFFFFFFFu),   /* global_addr[31:0]      */ \
                (unsigned)((gaddr >> 32) & 0x01FFFFFFu) | (2u << 30) };        \
            int32x8 g1 = {                                                     \
                (int)(2u << 16),                   /* data_size = 4B         */ \
                (int)((unsigned)D_FEAT << 16),     /* tensor_dim0 = 64       */ \
                (int)(((unsigned)N_NODES & 0xFFFFu) << 16), /* dim1 lo16     */ \
                (int)(((unsigned)N_NODES >> 16) |  /* dim1 hi16 | tile_dim0  */ \
                      ((unsigned)D_FEAT << 16)),                               \
                8,                                 /* tile_dim1 = #indices   */ \
                D_FEAT,                            /* dim0_stride = 64       */ \
                0, 0 };                                                        \
            int32x8 g4 = { 0, 0, 0, 0, 0, 0, 0, 0 }; /* unused ext group     */ \
            __builtin_amdgcn_tensor_load_to_lds(g0, g1, g2, g3, g4, 0);        \
        } while (0)

        // Consume 8 gathered rows from buffer BUF for edges [8K, 8K+8).
#define TDM_CONSUME(K, BUF)                                                    \
        do {                                                                   \
            _Pragma("unroll")                                                  \
            for (int j = 0; j < 8; ++j) {                                      \
                const int   ri = __builtin_amdgcn_readlane(r, (K)*8 + j);      \
                const float vi = __int_as_float(                               \
                    __builtin_amdgcn_readlane(__float_as_int(v), (K)*8 + j));  \
                const float2 emb = *(const float2*)                            \
                    &smem[slot + (BUF)*512 + j*64 + feat];                     \
                float* dst = out + (size_t)ri * D_FEAT + feat;                 \
                atomicAdd(dst + 0, vi * emb.x);                                \
                atomicAdd(dst + 1, vi * emb.y);                                \
            }                                                                  \
        } while (0)

#define TDM_WAIT(N)                                                            \
        do {                                                                   \
            __builtin_amdgcn_s_wait_tensorcnt((short)(N));                     \
            __builtin_amdgcn_sched_barrier(0);                                 \
            asm volatile("" ::: "memory");                                     \
        } while (0)

        TDM_ISSUE(0, 0);
        TDM_ISSUE(1, 1);
        TDM_WAIT(1);          // chunk 0 landed
        TDM_CONSUME(0, 0);
        TDM_ISSUE(2, 0);
        TDM_WAIT(1);          // chunk 1 landed
        TDM_CONSUME(1, 1);
        TDM_ISSUE(3, 1);
        TDM_WAIT(1);          // chunk 2 landed
        TDM_CONSUME(2, 0);
        TDM_WAIT(0);          // chunk 3 landed
        TDM_CONSUME(3, 1);

#undef TDM_ISSUE
#undef TDM_CONSUME
#undef TDM_WAIT
    } else {
        // ---- generic tail (not hit for E = 1.6M): direct gather ----
        const int n = E - base;
        for (int i = 0; i < n; ++i) {
            const int   ri = __shfl(r, i, 32);
            const int   ci = __shfl(c, i, 32);
            const float vi = __shfl(v, i, 32);
            const float2 emb = *(const float2*)(embeds + (size_t)ci * D_FEAT + feat);
            float* dst = out + (size_t)ri * D_FEAT + feat;
            atomicAdd(dst + 0, vi * emb.x);
            atomicAdd(dst + 1, vi * emb.y);
        }
    }
}

extern "C" void kernel_launch(void* const* d_in, const int* in_sizes, int n_in,
                              void* d_out, int out_size, void* d_ws, size_t ws_size,
                              hipStream_t stream) {
    (void)n_in; (void)d_ws; (void)ws_size;

    // Inputs in setup_inputs() order:
    //   d_in[0]: edge_index  int32 [2, E]  (rows then cols)
    //   d_in[1]: edge_vals   f32   [E]
    //   d_in[2]: embeds      f32   [N, 64]
    const int*   edge_index = (const int*)d_in[0];
    const float* edge_vals  = (const float*)d_in[1];
    const float* embeds     = (const float*)d_in[2];
    float*       out        = (float*)d_out;

    const int E     = in_sizes[0] / 2;    // 1,600,000
    const int* rows = edge_index;
    const int* cols = edge_index + E;

    // Zero output (N*D floats) with float4 stores.
    const int n4 = out_size / 4;
    zero_out_kernel<<<(n4 + 255) / 256, 256, 0, stream>>>((float4*)out, n4);

    // One wave per 32 edges; 8 waves/block; 32KB dynamic LDS (2x2KB per wave).
    const long long threads = ((long long)(E + 31) / 32) * 32;
    const int blocks = (int)((threads + 255) / 256);
    spmm_coo_scatter_tdm<<<blocks, 256, 32768, stream>>>(
        rows, cols, edge_vals, embeds, out, E);
}